// GCN_55972013801739
// MI455X (gfx1250) — compile-verified
//
#include <hip/hip_runtime.h>

typedef float v2f __attribute__((ext_vector_type(2)));
typedef float v8f __attribute__((ext_vector_type(8)));

#define GCN_N_NODES 100000
#define GCN_N_EDGES 1600000

// ---------------------------------------------------------------------------
// Dense GEMM: C[M,N] = A[M,K] @ W[N,K]^T (+ bias), via V_WMMA_F32_16X16X4_F32.
// One wave computes one 16x16 C tile. All control flow before the WMMA loop is
// wave-uniform so EXEC stays all-ones (WMMA requirement).
// A fragment (16x4 f32): lane holds row m = m0+(lane&15); lane bit4 selects the
// K pair (K=0,1 vs K=2,3)  -> one float2 load per step.
// B fragment (4x16): lane holds column n (from W row n, since B = W^T); same
// K-pair selection -> one float2 load per step.
// C/D layout: VGPR r -> (M = m0 + r + 8*lane[4], N = n0 + (lane&15)).
// ---------------------------------------------------------------------------
__global__ void gemm_nt_wmma_f32(const float* __restrict__ A,
                                 const float* __restrict__ W,
                                 const float* __restrict__ bias,
                                 float* __restrict__ C,
                                 int M, int N, int K, int ldc,
                                 int tilesM, int tilesN)
{
    const int lane = threadIdx.x & 31;
    const int wave = blockIdx.x * (blockDim.x >> 5) + (threadIdx.x >> 5);
    if (wave >= tilesM * tilesN) return;   // wave-uniform exit

    const int tm = wave % tilesM;
    const int tn = wave / tilesM;
    const int m0 = tm << 4;
    const int n0 = tn << 4;

    int mrow = m0 + (lane & 15); if (mrow >= M) mrow = M - 1;  // clamp (no OOB)
    int ncol = n0 + (lane & 15); if (ncol >= N) ncol = N - 1;
    const int kgrp = (lane >> 4) << 1;

    const float* arow = A + (size_t)mrow * K + kgrp;
    const float* wrow = W + (size_t)ncol * K + kgrp;

    v8f acc = {};
#pragma unroll 4
    for (int k = 0; k < K; k += 4) {
        v2f a = *(const v2f*)(arow + k);
        v2f b = *(const v2f*)(wrow + k);
        acc = __builtin_amdgcn_wmma_f32_16x16x4_f32(
            /*neg_a=*/false, a, /*neg_b=*/false, b,
            /*c_mod=*/(short)0, acc, /*reuse_a=*/false, /*reuse_b=*/false);
    }

    const int cn = n0 + (lane & 15);
    if (cn < N) {
        const float bv = bias ? bias[cn] : 0.0f;
        const int mbase = m0 + ((lane >> 4) << 3);
#pragma unroll
        for (int r = 0; r < 8; ++r) {
            const int cm = mbase + r;
            if (cm < M) C[(size_t)cm * ldc + cn] = acc[r] + bv;
        }
    }
}

// ---------------------------------------------------------------------------
// Degree / norm helpers
// ---------------------------------------------------------------------------
__global__ void set_ones_k(float* __restrict__ p, int n)
{
    int i = blockIdx.x * blockDim.x + threadIdx.x;
    if (i < n) p[i] = 1.0f;               // self loop contributes 1 to degree
}

__global__ void deg_accum_k(const int* __restrict__ dst, float* __restrict__ deg, int E)
{
    int e = blockIdx.x * blockDim.x + threadIdx.x;
    if (e < E) atomicAdd(&deg[dst[e]], 1.0f);
}

__global__ void rsqrt_k(float* __restrict__ dinv, const float* __restrict__ deg, int n)
{
    int i = blockIdx.x * blockDim.x + threadIdx.x;
    if (i < n) dinv[i] = rsqrtf(deg[i]);  // deg >= 1 always (self loop)
}

// out[i,f] = h[i,f] * dinv[i]^2   (self-loop message; also initializes out)
__global__ void self_init_k(const float* __restrict__ h,
                            const float* __restrict__ dinv,
                            float* __restrict__ out,
                            size_t total, int shift)
{
    size_t i = (size_t)blockIdx.x * blockDim.x + threadIdx.x;
    if (i < total) {
        float dv = dinv[i >> shift];
        out[i] = h[i] * dv * dv;
    }
}

// One wave per edge: lane owns F/32 contiguous features.
// Gather h[src] (coalesced 128b/64b), scale by dinv[s]*dinv[d], scatter-add
// into out[dst] with global_atomic_add_f32. Data is L2-resident on MI455X.
template <int F>
__global__ void edge_agg_k(const float* __restrict__ h,
                           const int* __restrict__ src,
                           const int* __restrict__ dst,
                           const float* __restrict__ dinv,
                           float* __restrict__ out, int E)
{
    constexpr int V = F / 32;
    const int wid = blockIdx.x * (blockDim.x >> 5) + (threadIdx.x >> 5);
    if (wid >= E) return;                 // wave-uniform
    const int s = src[wid];
    const int d = dst[wid];
    const float norm = dinv[s] * dinv[d];
    const int lane = threadIdx.x & 31;
    const float* hp = h + (size_t)s * F + lane * V;
    float* op = out + (size_t)d * F + lane * V;
    float vbuf[V];
#pragma unroll
    for (int v = 0; v < V; ++v) vbuf[v] = hp[v];
#pragma unroll
    for (int v = 0; v < V; ++v) atomicAdd(op + v, vbuf[v] * norm);
}

__global__ void bias_relu_k(float* __restrict__ xio, const float* __restrict__ b,
                            size_t total, int mask)
{
    size_t i = (size_t)blockIdx.x * blockDim.x + threadIdx.x;
    if (i < total) {
        float v = xio[i] + b[i & mask];
        xio[i] = v > 0.0f ? v : 0.0f;
    }
}

// ---------------------------------------------------------------------------
extern "C" void kernel_launch(void* const* d_in, const int* in_sizes, int n_in,
                              void* d_out, int out_size, void* d_ws, size_t ws_size,
                              hipStream_t stream)
{
    (void)in_sizes; (void)n_in; (void)out_size; (void)ws_size;

    const float* x   = (const float*)d_in[0];
    const int*   ei  = (const int*)d_in[1];     // [2, E]: row0 = src, row1 = dst
    const float* W1  = (const float*)d_in[2];   // [128,128]
    const float* b1  = (const float*)d_in[3];   // [128]
    const float* W2  = (const float*)d_in[4];   // [64,128]
    const float* b2  = (const float*)d_in[5];   // [64]
    const float* Wfc = (const float*)d_in[6];   // [40,64]
    const float* bfc = (const float*)d_in[7];   // [40]

    const int Nn = GCN_N_NODES;
    const int E  = GCN_N_EDGES;
    const int* src = ei;
    const int* dst = ei + E;

    // Workspace layout (floats): deg | dinv | h (N*128, reused for h2) | agg1 (N*128)
    float* ws   = (float*)d_ws;
    const size_t nAl = (size_t)((Nn + 255) & ~255);
    float* deg  = ws;
    float* dinv = ws + nAl;
    float* h    = ws + 2 * nAl;
    float* agg1 = h + (size_t)Nn * 128;

    float* emb    = (float*)d_out;              // [N,64]
    float* logits = emb + (size_t)Nn * 64;      // [N,40]

    const int TB = 256;
    const int wavesPerBlock = TB / 32;
    const int tilesM = (Nn + 15) / 16;          // 6250

    // --- degrees & symmetric norm ---
    set_ones_k<<<(Nn + TB - 1) / TB, TB, 0, stream>>>(deg, Nn);
    deg_accum_k<<<(E + TB - 1) / TB, TB, 0, stream>>>(dst, deg, E);
    rsqrt_k<<<(Nn + TB - 1) / TB, TB, 0, stream>>>(dinv, deg, Nn);

    // --- layer 1: h = x @ W1^T ; aggregate ; relu(+b1) ---
    {
        const int tilesN = 128 / 16;
        const int waves  = tilesM * tilesN;
        gemm_nt_wmma_f32<<<(waves + wavesPerBlock - 1) / wavesPerBlock, TB, 0, stream>>>(
            x, W1, nullptr, h, Nn, 128, 128, 128, tilesM, tilesN);
        const size_t tot = (size_t)Nn * 128;
        self_init_k<<<(unsigned)((tot + TB - 1) / TB), TB, 0, stream>>>(h, dinv, agg1, tot, 7);
        edge_agg_k<128><<<(E + wavesPerBlock - 1) / wavesPerBlock, TB, 0, stream>>>(
            h, src, dst, dinv, agg1, E);
        bias_relu_k<<<(unsigned)((tot + TB - 1) / TB), TB, 0, stream>>>(agg1, b1, tot, 127);
    }

    // --- layer 2: h2 = agg1 @ W2^T ; aggregate into d_out ; relu(+b2) ---
    {
        const int tilesN = 64 / 16;
        const int waves  = tilesM * tilesN;
        gemm_nt_wmma_f32<<<(waves + wavesPerBlock - 1) / wavesPerBlock, TB, 0, stream>>>(
            agg1, W2, nullptr, h, Nn, 64, 128, 64, tilesM, tilesN);
        const size_t tot = (size_t)Nn * 64;
        self_init_k<<<(unsigned)((tot + TB - 1) / TB), TB, 0, stream>>>(h, dinv, emb, tot, 6);
        edge_agg_k<64><<<(E + wavesPerBlock - 1) / wavesPerBlock, TB, 0, stream>>>(
            h, src, dst, dinv, emb, E);
        bias_relu_k<<<(unsigned)((tot + TB - 1) / TB), TB, 0, stream>>>(emb, b2, tot, 63);
    }

    // --- FC head: logits = emb @ Wfc^T + bfc (N=40 padded to 3 WMMA tiles) ---
    {
        const int tilesN = 3;                   // ceil(40/16)
        const int waves  = tilesM * tilesN;
        gemm_nt_wmma_f32<<<(waves + wavesPerBlock - 1) / wavesPerBlock, TB, 0, stream>>>(
            emb, Wfc, bfc, logits, Nn, 40, 64, 40, tilesM, tilesN);
    }
}